// SparseDecoder_21096879358070
// MI455X (gfx1250) — compile-verified
//
#include <hip/hip_runtime.h>
#include <hip/hip_bf16.h>

typedef __attribute__((ext_vector_type(2))) float v2f;
typedef __attribute__((ext_vector_type(8))) float v8f;
typedef __attribute__((ext_vector_type(4))) float f4v;
typedef __attribute__((ext_vector_type(4))) int   i4v;

#define BN_EPS 1e-5f

// ---------------------------------------------------------------------------
// CDNA5 async global->LDS copy (bypasses VGPRs, tracked by ASYNCcnt).
// VDST = 32-bit LDS byte address, VADDR = 64-bit global address, SADDR=off.
// ---------------------------------------------------------------------------
__device__ __forceinline__ void async_ld_b128(const float* gsrc, unsigned lds_byte) {
  asm volatile("global_load_async_to_lds_b128 %0, %1, off"
               :: "v"(lds_byte), "v"(gsrc)
               : "memory");
}
__device__ __forceinline__ void wait_async0() {
  asm volatile("s_wait_asynccnt 0" ::: "memory");
}
__device__ __forceinline__ unsigned lds_addr(const void* p) {
  // generic pointers into the LDS aperture carry the wave-relative LDS byte
  // offset in their low 32 bits (ISA 10.2 aperture mapping)
  return (unsigned)(uintptr_t)p;
}

// ============================================================================
// Dense-staged WMMA GEMM with double-buffered async LDS pipeline:
//   out(M,N) = A(M,K) @ Bw(N,K)^T + bias     (Bw already mask-multiplied)
// Block tile 128(M) x 64(N), K chunks of 32. 8 waves; wave w owns rows
// [w*16, w*16+16) x 64 cols as 4 accumulators via V_WMMA_F32_16X16X4_F32.
// ============================================================================
__global__ __launch_bounds__(256)
void gemm_wmma_async(const float* __restrict__ A,    // (M, K)
                     const float* __restrict__ Bw,   // (N, K)  (= W * Mask)
                     const float* __restrict__ bias, // (N)
                     float* __restrict__ out,        // (M, N)
                     int Ndim, int Kdim)
{
  // row stride 36 floats: 16B-aligned rows (144 % 16 == 0), 8B-aligned v2f
  // fragments, bank-spread across the 64 LDS banks. Two buffers for overlap.
  __shared__ float As[2][128][36];
  __shared__ float Bs[2][64][36];

  const int tid   = threadIdx.x;
  const int wave  = tid >> 5;
  const int lane  = tid & 31;
  const int lm    = lane & 15;          // row (A) / col (B,D) within tile
  const int khalf = (lane >> 4) << 1;   // lanes 0-15: K0/K1, lanes 16-31: K2/K3

  const int n0 = blockIdx.x * 64;
  const int m0 = blockIdx.y * 128;

  // pure-copy staging decomposition
  const int ar = tid >> 1;              // A row 0..127
  const int ah = (tid & 1) << 4;        // A col offset 0/16
  const int br = tid >> 2;              // B row 0..63
  const int bq = (tid & 3) << 3;        // B col offset 0,8,16,24

  const float* aSrc = A  + (size_t)(m0 + ar) * Kdim + ah;
  const float* bSrc = Bw + (size_t)(n0 + br) * Kdim + bq;

  auto stage = [&](int buf, int kt) {
    const float* as = aSrc + kt;
    const unsigned ad = lds_addr(&As[buf][ar][ah]);
    async_ld_b128(as,      ad);
    async_ld_b128(as + 4,  ad + 16);
    async_ld_b128(as + 8,  ad + 32);
    async_ld_b128(as + 12, ad + 48);
    const float* bs = bSrc + kt;
    const unsigned bd = lds_addr(&Bs[buf][br][bq]);
    async_ld_b128(bs,     bd);
    async_ld_b128(bs + 4, bd + 16);
  };

  v8f acc[4] = {};

  stage(0, 0);
  wait_async0();
  __syncthreads();

  int p = 0;
  for (int kt = 0; kt < Kdim; kt += 32) {
    // overlap: issue next chunk's async copies, then compute current chunk
    if (kt + 32 < Kdim) stage(p ^ 1, kt + 32);

#pragma unroll
    for (int ks = 0; ks < 8; ++ks) {
      const int kb = (ks << 2) + khalf;
      v2f a = *(const v2f*)&As[p][(wave << 4) + lm][kb];
#pragma unroll
      for (int nt = 0; nt < 4; ++nt) {
        v2f b = *(const v2f*)&Bs[p][(nt << 4) + lm][kb];
        acc[nt] = __builtin_amdgcn_wmma_f32_16x16x4_f32(
            false, a, false, b, (short)0, acc[nt], false, false);
      }
    }

    wait_async0();      // next chunk's copies finished during compute
    __syncthreads();
    p ^= 1;
  }

  // epilogue: D layout = VGPR v -> row v (+8 for upper lane half)
  const int col0  = n0 + lm;
  const int rbase = m0 + (wave << 4) + ((lane >> 4) << 3);
#pragma unroll
  for (int nt = 0; nt < 4; ++nt) {
    const int c = col0 + (nt << 4);
    const float bv = bias[c];
#pragma unroll
    for (int v = 0; v < 8; ++v) {
      out[(size_t)(rbase + v) * Ndim + c] = acc[nt][v] + bv;
    }
  }
}

// ============================================================================
// One-time prep: Weff = W * Mask (vectorized); makes GEMM staging a pure copy.
// ============================================================================
__global__ __launch_bounds__(256)
void mask_apply(const float* __restrict__ W, const int* __restrict__ M,
                float* __restrict__ Weff, int total4)
{
  const int i = blockIdx.x * 256 + threadIdx.x;
  if (i >= total4) return;
  f4v w = ((const f4v*)W)[i];
  i4v m = ((const i4v*)M)[i];
  f4v v;
#pragma unroll
  for (int j = 0; j < 4; ++j) v[j] = m[j] ? w[j] : 0.f;
  ((f4v*)Weff)[i] = v;
}

// ============================================================================
// BatchNorm column statistics (training mode, biased variance), two stages.
// ============================================================================
__global__ __launch_bounds__(256)
void col_stats_partial(const float* __restrict__ h,
                       float* __restrict__ psum, float* __restrict__ psq,
                       int Mrows, int D)
{
  const int c  = (blockIdx.x << 6) + (threadIdx.x & 63);
  const int rl = threadIdx.x >> 6;
  const int rows = Mrows / gridDim.y;
  const int r0 = blockIdx.y * rows;
  float s = 0.f, q = 0.f;
  for (int r = r0 + rl; r < r0 + rows; r += 4) {
    float v = h[(size_t)r * D + c];
    s += v; q += v * v;
  }
  __shared__ float ls[4][64];
  __shared__ float lq[4][64];
  ls[rl][threadIdx.x & 63] = s;
  lq[rl][threadIdx.x & 63] = q;
  __syncthreads();
  if (rl == 0) {
    const int t = threadIdx.x & 63;
    s = ls[0][t] + ls[1][t] + ls[2][t] + ls[3][t];
    q = lq[0][t] + lq[1][t] + lq[2][t] + lq[3][t];
    psum[(size_t)blockIdx.y * D + c] = s;
    psq [(size_t)blockIdx.y * D + c] = q;
  }
}

__global__ __launch_bounds__(256)
void col_stats_final(const float* __restrict__ psum, const float* __restrict__ psq,
                     const float* __restrict__ gamma, const float* __restrict__ beta,
                     float* __restrict__ scale, float* __restrict__ shift,
                     int D, int nchunks, float invM)
{
  const int c = blockIdx.x * 256 + threadIdx.x;
  if (c >= D) return;
  float s = 0.f, q = 0.f;
  for (int i = 0; i < nchunks; ++i) {
    s += psum[(size_t)i * D + c];
    q += psq [(size_t)i * D + c];
  }
  const float mean = s * invM;
  const float var  = q * invM - mean * mean;
  const float sc   = gamma[c] * rsqrtf(var + BN_EPS);
  scale[c] = sc;
  shift[c] = beta[c] - mean * sc;
}

// ============================================================================
// In-place BN+ReLU on the pre-BN activations (h -> relu(h*scale+shift)).
// ============================================================================
__global__ __launch_bounds__(256)
void bn_relu_inplace(float* __restrict__ h, const float* __restrict__ scale,
                     const float* __restrict__ shift, int D, int total4)
{
  const int i = blockIdx.x * 256 + threadIdx.x;   // float4 index
  if (i >= total4) return;
  const int col = (i << 2) % D;                   // D is a multiple of 4
  f4v v = ((f4v*)h)[i];
#pragma unroll
  for (int j = 0; j < 4; ++j) {
    float t = v[j] * scale[col + j] + shift[col + j];
    v[j] = t > 0.f ? t : 0.f;
  }
  ((f4v*)h)[i] = v;
}

// ============================================================================
// Launcher: mask-prep x3, then GEMM -> stats -> BN+ReLU -> GEMM -> ... -> out
// ============================================================================
extern "C" void kernel_launch(void* const* d_in, const int* in_sizes, int n_in,
                              void* d_out, int out_size, void* d_ws, size_t ws_size,
                              hipStream_t stream) {
  const float* x   = (const float*)d_in[0];
  const float* W1  = (const float*)d_in[1];
  const float* b1  = (const float*)d_in[2];
  const int*   M1  = (const int*)  d_in[3];
  const float* g1  = (const float*)d_in[4];
  const float* be1 = (const float*)d_in[5];
  const float* W2  = (const float*)d_in[6];
  const float* b2  = (const float*)d_in[7];
  const int*   M2  = (const int*)  d_in[8];
  const float* g2  = (const float*)d_in[9];
  const float* be2 = (const float*)d_in[10];
  const float* W3  = (const float*)d_in[11];
  const float* b3  = (const float*)d_in[12];
  const int*   M3  = (const int*)  d_in[13];

  const int D1 = in_sizes[2];            // 1024
  const int D2 = in_sizes[7];            // 2048
  const int D3 = in_sizes[12];           // 4096
  const int D0 = in_sizes[1] / D1;       // 256
  const int B  = in_sizes[0] / D0;       // 8192
  const int NCHUNK = 16;

  float* ws = (float*)d_ws;
  float* h1    = ws; ws += (size_t)B * D1;        // pre-BN acts (BN'd in place)
  float* h2    = ws; ws += (size_t)B * D2;
  float* We1   = ws; ws += (size_t)D1 * D0;       // masked weights
  float* We2   = ws; ws += (size_t)D2 * D1;
  float* We3   = ws; ws += (size_t)D3 * D2;
  float* psum  = ws; ws += (size_t)NCHUNK * D2;
  float* psq   = ws; ws += (size_t)NCHUNK * D2;
  float* scale1 = ws; ws += D1;
  float* shift1 = ws; ws += D1;
  float* scale2 = ws; ws += D2;
  float* shift2 = ws; ws += D2;

  float* outp = (float*)d_out;
  dim3 blk(256);

  // one-time masked-weight materialization (pure-copy staging downstream)
  const int t1 = D1 * D0 / 4, t2 = D2 * D1 / 4, t3 = D3 * D2 / 4;
  mask_apply<<<dim3((t1 + 255) / 256), blk, 0, stream>>>(W1, M1, We1, t1);
  mask_apply<<<dim3((t2 + 255) / 256), blk, 0, stream>>>(W2, M2, We2, t2);
  mask_apply<<<dim3((t3 + 255) / 256), blk, 0, stream>>>(W3, M3, We3, t3);

  // Layer 1: h1 = x @ We1^T + b1
  gemm_wmma_async<<<dim3(D1 / 64, B / 128), blk, 0, stream>>>(x, We1, b1, h1, D1, D0);
  col_stats_partial<<<dim3(D1 / 64, NCHUNK), blk, 0, stream>>>(h1, psum, psq, B, D1);
  col_stats_final<<<dim3((D1 + 255) / 256), blk, 0, stream>>>(
      psum, psq, g1, be1, scale1, shift1, D1, NCHUNK, 1.0f / (float)B);
  {
    const int ta = B * D1 / 4;
    bn_relu_inplace<<<dim3((ta + 255) / 256), blk, 0, stream>>>(h1, scale1, shift1, D1, ta);
  }

  // Layer 2: h2 = a1 @ We2^T + b2
  gemm_wmma_async<<<dim3(D2 / 64, B / 128), blk, 0, stream>>>(h1, We2, b2, h2, D2, D1);
  col_stats_partial<<<dim3(D2 / 64, NCHUNK), blk, 0, stream>>>(h2, psum, psq, B, D2);
  col_stats_final<<<dim3((D2 + 255) / 256), blk, 0, stream>>>(
      psum, psq, g2, be2, scale2, shift2, D2, NCHUNK, 1.0f / (float)B);
  {
    const int ta = B * D2 / 4;
    bn_relu_inplace<<<dim3((ta + 255) / 256), blk, 0, stream>>>(h2, scale2, shift2, D2, ta);
  }

  // Layer 3: out = a2 @ We3^T + b3
  gemm_wmma_async<<<dim3(D3 / 64, B / 128), blk, 0, stream>>>(h2, We3, b3, outp, D3, D2);
}